// PQMatcher_6725918785796
// MI455X (gfx1250) — compile-verified
//
#include <hip/hip_runtime.h>
#include <hip/hip_bf16.h>
#include <math.h>

// Problem sizes (from reference)
#define B_SZ   24
#define LP_SZ  512
#define LQ_SZ  512
#define IN_SZ  150
#define H_SZ   150
#define HP_SZ  160   // h padded to 10 WMMA tiles of 16
#define KP_SZ  152   // k padded to multiple of 4

typedef float v2f __attribute__((ext_vector_type(2)));
typedef float v8f __attribute__((ext_vector_type(8)));

// ---- gfx1250 fast-math helpers (hardware trans ops, guarded) --------------
__device__ __forceinline__ float fast_tanhf(float x) {
#if __has_builtin(__builtin_amdgcn_tanhf)
  return __builtin_amdgcn_tanhf(x);   // v_tanh_f32 (TRANS32)
#else
  return tanhf(x);
#endif
}
__device__ __forceinline__ float fast_expf(float x) {
#if __has_builtin(__builtin_amdgcn_exp2f)
  return __builtin_amdgcn_exp2f(x * 1.44269504088896340736f);  // v_exp_f32
#else
  return __expf(x);
#endif
}
__device__ __forceinline__ float fast_rcpf(float x) {
#if __has_builtin(__builtin_amdgcn_rcpf)
  return __builtin_amdgcn_rcpf(x);    // v_rcp_f32
#else
  return 1.0f / x;
#endif
}
__device__ __forceinline__ float sigmoidf(float x) {
  return fast_rcpf(1.0f + fast_expf(-x));
}

__device__ __forceinline__ float wave_sum(float v) {
#pragma unroll
  for (int off = 16; off > 0; off >>= 1) v += __shfl_xor(v, off, 32);
  return v;
}
__device__ __forceinline__ float wave_max(float v) {
#pragma unroll
  for (int off = 16; off > 0; off >>= 1) v = fmaxf(v, __shfl_xor(v, off, 32));
  return v;
}

// ---------------------------------------------------------------------------
// Kernel 1: fold the concat([u,u]) structure into the weights.
//   WqP/WpP : [HP_SZ][KP_SZ] zero-padded, W_eff[h][d] = W[h][d] + W[h][150+d]
//   Wg_a    : [600][150]  = Wg[:, 0:150]   + Wg[:, 150:300]   (acts on up_i)
//   Wg_c    : [600][150]  = Wg[:, 300:450] + Wg[:, 450:600]   (acts on ch)
// ---------------------------------------------------------------------------
__global__ void prep_kernel(const float* __restrict__ Wp,
                            const float* __restrict__ Wq,
                            const float* __restrict__ Wg,
                            float* __restrict__ WpP, float* __restrict__ WqP,
                            float* __restrict__ Wg_a, float* __restrict__ Wg_c) {
  int idx = blockIdx.x * blockDim.x + threadIdx.x;
  int stride = gridDim.x * blockDim.x;
  const int tot1 = HP_SZ * KP_SZ;
  for (int i = idx; i < tot1; i += stride) {
    int h = i / KP_SZ, d = i % KP_SZ;
    float vq = 0.f, vp = 0.f;
    if (h < H_SZ && d < IN_SZ) {
      vq = Wq[h * 300 + d] + Wq[h * 300 + 150 + d];
      vp = Wp[h * 300 + d] + Wp[h * 300 + 150 + d];
    }
    WqP[i] = vq;
    WpP[i] = vp;
  }
  const int tot2 = 600 * IN_SZ;
  for (int i = idx; i < tot2; i += stride) {
    int j = i / IN_SZ, d = i % IN_SZ;
    Wg_a[i] = Wg[j * 600 + d] + Wg[j * 600 + 150 + d];
    Wg_c[i] = Wg[j * 600 + 300 + d] + Wg[j * 600 + 450 + d];
  }
}

// ---------------------------------------------------------------------------
// Kernel 2: transpose uq [L][B][IN] -> uqT [B][L][IN] (b-major for the scan)
// ---------------------------------------------------------------------------
__global__ void tr_kernel(const float* __restrict__ uq, float* __restrict__ uqT) {
  int row = blockIdx.x;            // row = l*B + b
  int l = row / B_SZ, b = row % B_SZ;
  for (int d = threadIdx.x; d < IN_SZ; d += blockDim.x)
    uqT[((size_t)b * LQ_SZ + l) * IN_SZ + d] = uq[(size_t)row * IN_SZ + d];
}

// ---------------------------------------------------------------------------
// Kernel 3: WMMA f32 GEMM, transposed output.
//   Y[b][l][h] = sum_d X[l][b][d] * W[h][d]      (X: [512][24][150], W padded)
// One wave per 16(l)x16(h) tile row, 10 h-tiles of 16, K loop in steps of 4
// using V_WMMA_F32_16X16X4_F32.  A layout: lane m=L%16 holds row l0+m,
// VGPR pair = K (k0+2*half, +1).  B layout: lane n=L%16 holds col h0+n, same K
// pair.  C layout: reg r, lanes0-15 -> (M=r, N=lane); lanes16-31 -> (M=r+8).
// ---------------------------------------------------------------------------
__global__ __launch_bounds__(256) void gemm_t_kernel(const float* __restrict__ X,
                                                     const float* __restrict__ W,
                                                     float* __restrict__ Y) {
  int wv = threadIdx.x >> 5;
  int lane = threadIdx.x & 31;
  int tile = blockIdx.x * 8 + wv;              // 24 b * 32 l-tiles = 768 tiles
  int b = tile >> 5;
  int l0 = (tile & 31) * 16;
  int m = lane & 15;
  int half = lane >> 4;

  v8f acc[10];
#pragma unroll
  for (int nt = 0; nt < 10; ++nt) acc[nt] = (v8f)0.f;

  for (int k0 = 0; k0 < KP_SZ; k0 += 4) {
    int d = k0 + 2 * half;
    v2f a;
    if (d < IN_SZ) {
      a = *(const v2f*)(X + ((size_t)(l0 + m) * B_SZ + b) * IN_SZ + d);
    } else {
      a.x = 0.f; a.y = 0.f;
    }
#pragma unroll
    for (int nt = 0; nt < 10; ++nt) {
      int n = nt * 16 + m;
      v2f bb = *(const v2f*)(W + (size_t)n * KP_SZ + d);   // padded, in-bounds
#if __has_builtin(__builtin_amdgcn_wmma_f32_16x16x4_f32)
      acc[nt] = __builtin_amdgcn_wmma_f32_16x16x4_f32(
          false, a, false, bb, (short)0, acc[nt], false, false);
#else
      // scalar fallback keeping identical C layout
#pragma unroll
      for (int r = 0; r < 8; ++r) {
        int l = l0 + r + 8 * half;
        int h = nt * 16 + m;
        float s = 0.f;
        for (int kk = 0; kk < 4; ++kk) {
          int dd = k0 + kk;
          if (dd < IN_SZ)
            s += X[((size_t)l * B_SZ + b) * IN_SZ + dd] * W[(size_t)h * KP_SZ + dd];
        }
        acc[nt][r] += s;
      }
#endif
    }
  }

#pragma unroll
  for (int nt = 0; nt < 10; ++nt) {
#pragma unroll
    for (int r = 0; r < 8; ++r) {
      int l = l0 + r + 8 * half;
      int h = nt * 16 + m;
      Y[((size_t)b * LP_SZ + l) * HP_SZ + h] = acc[nt][r];
    }
  }
}

// ---------------------------------------------------------------------------
// Kernel 4: persistent per-b scan. One workgroup (8 waves) per batch row b;
// the entire 512-step recurrence runs in LDS with only workgroup barriers.
// Uses hardware v_tanh_f32 / v_exp_f32 / v_rcp_f32 and global_prefetch_b8.
// ---------------------------------------------------------------------------
__global__ __launch_bounds__(256) void scan_kernel(
    const float* __restrict__ WUqT,  // [B][LQ][HP]
    const float* __restrict__ WupT,  // [B][LP][HP]
    const float* __restrict__ uqT,   // [B][LQ][IN]
    const float* __restrict__ up,    // [LP][B][IN]
    const float* __restrict__ V,     // [B][H]
    const float* __restrict__ v0,    // [B][H]
    const float* __restrict__ Wv,    // [H][H]
    const float* __restrict__ Wg_a,  // [600][IN]
    const float* __restrict__ Wg_c,  // [600][IN]
    const float* __restrict__ w_ih,  // [450][300]
    const float* __restrict__ w_hh,  // [450][H]
    const float* __restrict__ b_ih,  // [450]
    const float* __restrict__ b_hh,  // [450]
    float* __restrict__ out) {       // [LP][B][H]
  __shared__ float v_s[H_SZ], Vv_s[H_SZ], WuW_s[H_SZ], upi_s[IN_SZ], ch_s[IN_SZ];
  __shared__ float s_s[LQ_SZ], g_s[600], cb_s[300], gi_s[450], gh_s[450], red_s[8];

  const int b = blockIdx.x;
  const int t = threadIdx.x;
  const int wv = t >> 5;
  const int lane = t & 31;

  for (int h = t; h < H_SZ; h += 256) {
    v_s[h] = v0[b * H_SZ + h];
    Vv_s[h] = V[b * H_SZ + h];          // V[:, :, 0]
  }
  __syncthreads();

  const float* WUqb = WUqT + (size_t)b * LQ_SZ * HP_SZ;
  const float* uqb = uqT + (size_t)b * LQ_SZ * IN_SZ;

  for (int i = 0; i < LP_SZ; ++i) {
    // --- prefetch next step's serial-path streams into the cache ----------
    if (wv == 0 && lane < 5 && i + 1 < LP_SZ) {
      __builtin_prefetch(
          (const char*)(up + ((size_t)(i + 1) * B_SZ + b) * IN_SZ) + lane * 128, 0, 0);
      __builtin_prefetch(
          (const char*)(WupT + ((size_t)b * LP_SZ + i + 1) * HP_SZ) + lane * 128, 0, 0);
    }
    // --- load Up_i row, compute WuW[h] = (v @ Wv^T)[h] + Wup_i[h] ----------
    for (int d = t; d < IN_SZ; d += 256)
      upi_s[d] = up[((size_t)i * B_SZ + b) * IN_SZ + d];
    for (int h = wv; h < H_SZ; h += 8) {
      float p = 0.f;
      for (int d = lane; d < H_SZ; d += 32) p += Wv[h * H_SZ + d] * v_s[d];
      p = wave_sum(p);
      if (lane == 0)
        WuW_s[h] = p + WupT[((size_t)b * LP_SZ + i) * HP_SZ + h];
    }
    __syncthreads();

    // --- scores s[l] = sum_h tanh(WUq[l,h] + WuW[h]) * Vv[h] ---------------
    for (int l = wv; l < LQ_SZ; l += 8) {
      const float* row = WUqb + (size_t)l * HP_SZ;
      float p = 0.f;
      for (int h = lane; h < H_SZ; h += 32)
        p += fast_tanhf(row[h] + WuW_s[h]) * Vv_s[h];
      p = wave_sum(p);
      if (lane == 0) s_s[l] = p;
    }
    __syncthreads();

    // --- softmax over l (in place in s_s) ----------------------------------
    float mx = -3.4e38f;
    for (int l = t; l < LQ_SZ; l += 256) mx = fmaxf(mx, s_s[l]);
    mx = wave_max(mx);
    if (lane == 0) red_s[wv] = mx;
    __syncthreads();
    mx = red_s[0];
#pragma unroll
    for (int k = 1; k < 8; ++k) mx = fmaxf(mx, red_s[k]);
    __syncthreads();  // red_s reuse
    float sm = 0.f;
    for (int l = t; l < LQ_SZ; l += 256) {
      float e = fast_expf(s_s[l] - mx);
      s_s[l] = e;
      sm += e;
    }
    sm = wave_sum(sm);
    if (lane == 0) red_s[wv] = sm;
    __syncthreads();
    sm = red_s[0];
#pragma unroll
    for (int k = 1; k < 8; ++k) sm += red_s[k];
    float inv = fast_rcpf(sm);
    for (int l = t; l < LQ_SZ; l += 256) s_s[l] *= inv;
    // zero ch accumulator
    for (int d = t; d < IN_SZ; d += 256) ch_s[d] = 0.f;
    __syncthreads();

    // --- ch[d] = sum_l a[l] * uq[b,l,d]  (wave-partials + LDS f32 atomics) -
    {
      float pch[5] = {0.f, 0.f, 0.f, 0.f, 0.f};
      for (int l = wv * 64; l < wv * 64 + 64; ++l) {
        float a = s_s[l];
        const float* rowu = uqb + (size_t)l * IN_SZ;
#pragma unroll
        for (int k = 0; k < 5; ++k) {
          int d = lane + 32 * k;
          if (d < IN_SZ) pch[k] += a * rowu[d];
        }
      }
#pragma unroll
      for (int k = 0; k < 5; ++k) {
        int d = lane + 32 * k;
        if (d < IN_SZ) atomicAdd(&ch_s[d], pch[k]);   // ds_add_f32
      }
    }
    __syncthreads();

    // --- gates g[j] = sigmoid(up_i . Wg_a[j] + ch . Wg_c[j]) ---------------
    for (int j = wv; j < 600; j += 8) {
      float p = 0.f;
      for (int d = lane; d < IN_SZ; d += 32)
        p += upi_s[d] * Wg_a[j * IN_SZ + d] + ch_s[d] * Wg_c[j * IN_SZ + d];
      p = wave_sum(p);
      if (lane == 0) g_s[j] = sigmoidf(p);
    }
    __syncthreads();

    // --- c_ = (g*r)[:, 300:600] --------------------------------------------
    for (int d = t; d < IN_SZ; d += 256) {
      cb_s[d] = g_s[300 + d] * ch_s[d];
      cb_s[IN_SZ + d] = g_s[450 + d] * ch_s[d];
    }
    __syncthreads();

    // --- GRU pre-activations ----------------------------------------------
    for (int j = wv; j < 450; j += 8) {
      float pi = 0.f, ph = 0.f;
      for (int k = lane; k < 300; k += 32) pi += w_ih[j * 300 + k] * cb_s[k];
      for (int d = lane; d < H_SZ; d += 32) ph += w_hh[j * H_SZ + d] * v_s[d];
      pi = wave_sum(pi);
      ph = wave_sum(ph);
      if (lane == 0) {
        gi_s[j] = pi + b_ih[j];
        gh_s[j] = ph + b_hh[j];
      }
    }
    __syncthreads();

    // --- GRU cell update ---------------------------------------------------
    if (t < H_SZ) {
      float rr = sigmoidf(gi_s[t] + gh_s[t]);
      float zz = sigmoidf(gi_s[H_SZ + t] + gh_s[H_SZ + t]);
      float nn = fast_tanhf(gi_s[2 * H_SZ + t] + rr * gh_s[2 * H_SZ + t]);
      float vn = (1.f - zz) * nn + zz * v_s[t];
      out[((size_t)i * B_SZ + b) * H_SZ + t] = vn;
      v_s[t] = vn;
    }
    __syncthreads();
  }
}

// ---------------------------------------------------------------------------
extern "C" void kernel_launch(void* const* d_in, const int* in_sizes, int n_in,
                              void* d_out, int out_size, void* d_ws, size_t ws_size,
                              hipStream_t stream) {
  const float* up   = (const float*)d_in[0];
  const float* uq   = (const float*)d_in[1];
  const float* v0   = (const float*)d_in[2];
  const float* V    = (const float*)d_in[3];
  const float* Wp   = (const float*)d_in[4];
  const float* Wq   = (const float*)d_in[5];
  const float* Wv   = (const float*)d_in[6];
  const float* Wg   = (const float*)d_in[7];
  const float* w_ih = (const float*)d_in[8];
  const float* w_hh = (const float*)d_in[9];
  const float* b_ih = (const float*)d_in[10];
  const float* b_hh = (const float*)d_in[11];
  float* out = (float*)d_out;

  float* ws   = (float*)d_ws;
  float* WqP  = ws;                                   // 160*152
  float* WpP  = WqP + HP_SZ * KP_SZ;                  // 160*152
  float* Wg_a = WpP + HP_SZ * KP_SZ;                  // 600*150
  float* Wg_c = Wg_a + 600 * IN_SZ;                   // 600*150
  float* WUqT = Wg_c + 600 * IN_SZ;                   // 24*512*160
  float* WupT = WUqT + (size_t)B_SZ * LQ_SZ * HP_SZ;  // 24*512*160
  float* uqT  = WupT + (size_t)B_SZ * LP_SZ * HP_SZ;  // 24*512*150

  prep_kernel<<<64, 256, 0, stream>>>(Wp, Wq, Wg, WpP, WqP, Wg_a, Wg_c);
  tr_kernel<<<LQ_SZ * B_SZ, 256, 0, stream>>>(uq, uqT);
  gemm_t_kernel<<<96, 256, 0, stream>>>(uq, WqP, WUqT);  // WUqT
  gemm_t_kernel<<<96, 256, 0, stream>>>(up, WpP, WupT);  // WupT (all steps)
  scan_kernel<<<B_SZ, 256, 0, stream>>>(WUqT, WupT, uqT, up, V, v0, Wv, Wg_a,
                                        Wg_c, w_ih, w_hh, b_ih, b_hh, out);
}